// AFNO2D_42116449304746
// MI455X (gfx1250) — compile-verified
//
#include <hip/hip_runtime.h>
#include <hip/hip_bf16.h>

// AFNO2D forward for MI455X (gfx1250), all-f32 via V_WMMA_F32_16X16X4_F32.
// dht2d is factorized into 64x64 matmuls: stage1 P=C_hat*X, Q=S_hat*X (masked
// rows >=33 implement the rfft zero-pad), stage2 out = a*(K1*P + K2*Q) with
// K1=cos+sin (cas), K2=cos-sin. All mixing/bias/relu/shrink/residual are fused
// into the stage-2 epilogue. Tiles are staged into LDS with gfx1250
// GLOBAL_LOAD_ASYNC_TO_LDS_B128 (ASYNCcnt) to skip VGPR staging.

typedef float v2f __attribute__((ext_vector_type(2)));
typedef float v8f __attribute__((ext_vector_type(8)));

#define MST 68   // LDS row stride for transform matrices: 272B = 16B-aligned rows,
                 // column reads hit banks (4u+k)%64 -> conflict-free.

// async global->LDS copy of 16 bytes (per-lane). LDS byte address = low 32 bits
// of the flat pointer (generic->LDS truncation per CDNA5 aperture rules).
__device__ __forceinline__ void async_b128(const float* g, const float* l) {
  unsigned lds = (unsigned)(size_t)l;
  asm volatile("global_load_async_to_lds_b128 %0, %1, off" :: "v"(lds), "v"(g) : "memory");
}
__device__ __forceinline__ void async_wait0() {
  asm volatile("s_wait_asynccnt 0" ::: "memory");
}

// ---------------- basis matrices ----------------
__global__ void afno_basis(float* __restrict__ chat, float* __restrict__ shat,
                           float* __restrict__ k1,   float* __restrict__ k2) {
  int i = blockIdx.x * blockDim.x + threadIdx.x;   // 0..4095
  int u = i >> 6, t = i & 63;
  int p = (u * t) & 63;
  float ang = (float)p * 0.09817477042468103f;     // 2*pi/64
  float c = cosf(ang), s = sinf(ang);
  chat[i] = (u <= 32) ? c : 0.0f;                  // rfft keeps freqs 0..32 on axis 2
  shat[i] = (u <= 32) ? s : 0.0f;
  k1[i] = c + s;                                   // cas
  k2[i] = c - s;                                   // cas(-)
}

// ---------------- stage 1: P = Mc * X, Q = Ms * X (shared X tile) ----------------
// OUT[n,u,c] = sum_t M[u,t] * IN[(n/nD)*s1 + (n%nD)*s0 + t*st + c*sc]
__global__ __launch_bounds__(256)
void afno_g1(const float* __restrict__ in, float* __restrict__ outP, float* __restrict__ outQ,
             const float* __restrict__ Mc, const float* __restrict__ Ms,
             long nD, long s1, long s0, long st, long sc,
             long oD, long os1, long os0, long ost, long osc)
{
  __shared__ float Xl[64 * 64];
  __shared__ float McL[64 * MST];
  __shared__ float MsL[64 * MST];
  const int tid = threadIdx.x;
  const long n = blockIdx.x;
  const long cbase = (long)blockIdx.y * 64;
  const long ibase = (n / nD) * s1 + (n % nD) * s0;
  const long obase = (n / oD) * os1 + (n % oD) * os0;

  for (int i = tid * 4; i < 4096; i += 1024) {
    int r = i >> 6, k = i & 63;
    async_b128(Mc + i, &McL[r * MST + k]);
    async_b128(Ms + i, &MsL[r * MST + k]);
  }
  if (sc == 1) {
    for (int i = tid * 4; i < 4096; i += 1024) {
      int t = i >> 6, c = i & 63;
      async_b128(in + ibase + (long)t * st + (cbase + c), &Xl[t * 64 + c]);
    }
  } else {
    for (int i = tid; i < 4096; i += 256) {
      int t = i >> 6, c = i & 63;
      Xl[t * 64 + c] = in[ibase + (long)t * st + (cbase + c) * sc];
    }
  }
  async_wait0();
  __syncthreads();

  const int wv = tid >> 5, lane = tid & 31;
  const int half = lane >> 4, l15 = lane & 15;
  const int row0 = (wv >> 1) * 16;      // output-frequency block
  const int c0 = (wv & 1) * 32;         // pair of 16-wide channel blocks
  v8f aP0 = {}, aP1 = {}, aQ0 = {}, aQ1 = {};
  for (int kk = 0; kk < 64; kk += 4) {
    const int ka = kk + half * 2;
    v2f aC, aS, b0, b1;
    aC.x = McL[(row0 + l15) * MST + ka];  aC.y = McL[(row0 + l15) * MST + ka + 1];
    aS.x = MsL[(row0 + l15) * MST + ka];  aS.y = MsL[(row0 + l15) * MST + ka + 1];
    b0.x = Xl[ka * 64 + c0 + l15];       b0.y = Xl[(ka + 1) * 64 + c0 + l15];
    b1.x = Xl[ka * 64 + c0 + 16 + l15];  b1.y = Xl[(ka + 1) * 64 + c0 + 16 + l15];
    aP0 = __builtin_amdgcn_wmma_f32_16x16x4_f32(false, aC, false, b0, (short)0, aP0, false, false);
    aP1 = __builtin_amdgcn_wmma_f32_16x16x4_f32(false, aC, false, b1, (short)0, aP1, false, false);
    aQ0 = __builtin_amdgcn_wmma_f32_16x16x4_f32(false, aS, false, b0, (short)0, aQ0, false, false);
    aQ1 = __builtin_amdgcn_wmma_f32_16x16x4_f32(false, aS, false, b1, (short)0, aQ1, false, false);
  }
  for (int r = 0; r < 8; ++r) {
    const int u = row0 + r + half * 8;
    const long o = obase + (long)u * ost;
    outP[o + (cbase + c0 + l15) * osc]      = aP0[r];
    outP[o + (cbase + c0 + 16 + l15) * osc] = aP1[r];
    outQ[o + (cbase + c0 + l15) * osc]      = aQ0[r];
    outQ[o + (cbase + c0 + 16 + l15) * osc] = aQ1[r];
  }
}

// ---------------- stage 2: out = epilogue( alpha*(M1*P + M2*Q) ) ----------------
// modes: 0 none | 1 Hartley mix | 2 bias+relu | 3 bias+softshrink | 4 residual add
__global__ __launch_bounds__(256)
void afno_g2(const float* __restrict__ inP, const float* __restrict__ inQ,
             float* __restrict__ out,
             const float* __restrict__ M1, const float* __restrict__ M2,
             long nD, long s1, long s0, long st, long sc,
             long oD, long os1, long os0, long ost, long osc,
             float alpha, int mode,
             const float* __restrict__ bias,
             const float* __restrict__ xin,
             const float* __restrict__ ypl, const float* __restrict__ ymi,
             float lam)
{
  __shared__ float Pl[64 * 64];
  __shared__ float Ql[64 * 64];
  __shared__ float M1L[64 * MST];
  __shared__ float M2L[64 * MST];
  const int tid = threadIdx.x;
  const long n = blockIdx.x;
  const long cbase = (long)blockIdx.y * 64;
  const long ibase = (n / nD) * s1 + (n % nD) * s0;
  const long obase = (n / oD) * os1 + (n % oD) * os0;
  const int v = (int)(n % nD);   // axis-2 frequency index (for mode 1)

  for (int i = tid * 4; i < 4096; i += 1024) {
    int r = i >> 6, k = i & 63;
    async_b128(M1 + i, &M1L[r * MST + k]);
    async_b128(M2 + i, &M2L[r * MST + k]);
  }
  if (sc == 1) {
    for (int i = tid * 4; i < 4096; i += 1024) {
      int t = i >> 6, c = i & 63;
      long a = ibase + (long)t * st + (cbase + c);
      async_b128(inP + a, &Pl[t * 64 + c]);
      async_b128(inQ + a, &Ql[t * 64 + c]);
    }
  } else {
    for (int i = tid; i < 4096; i += 256) {
      int t = i >> 6, c = i & 63;
      long a = ibase + (long)t * st + (cbase + c) * sc;
      Pl[t * 64 + c] = inP[a];
      Ql[t * 64 + c] = inQ[a];
    }
  }
  async_wait0();
  __syncthreads();

  const int wv = tid >> 5, lane = tid & 31;
  const int half = lane >> 4, l15 = lane & 15;
  const int row0 = (wv >> 1) * 16;
  const int c0 = (wv & 1) * 32;
  v8f acc0 = {}, acc1 = {};
  for (int kk = 0; kk < 64; kk += 4) {
    const int ka = kk + half * 2;
    v2f a1, a2, bp0, bp1, bq0, bq1;
    a1.x = M1L[(row0 + l15) * MST + ka];  a1.y = M1L[(row0 + l15) * MST + ka + 1];
    a2.x = M2L[(row0 + l15) * MST + ka];  a2.y = M2L[(row0 + l15) * MST + ka + 1];
    bp0.x = Pl[ka * 64 + c0 + l15];      bp0.y = Pl[(ka + 1) * 64 + c0 + l15];
    bp1.x = Pl[ka * 64 + c0 + 16 + l15]; bp1.y = Pl[(ka + 1) * 64 + c0 + 16 + l15];
    bq0.x = Ql[ka * 64 + c0 + l15];      bq0.y = Ql[(ka + 1) * 64 + c0 + l15];
    bq1.x = Ql[ka * 64 + c0 + 16 + l15]; bq1.y = Ql[(ka + 1) * 64 + c0 + 16 + l15];
    acc0 = __builtin_amdgcn_wmma_f32_16x16x4_f32(false, a1, false, bp0, (short)0, acc0, false, false);
    acc0 = __builtin_amdgcn_wmma_f32_16x16x4_f32(false, a2, false, bq0, (short)0, acc0, false, false);
    acc1 = __builtin_amdgcn_wmma_f32_16x16x4_f32(false, a1, false, bp1, (short)0, acc1, false, false);
    acc1 = __builtin_amdgcn_wmma_f32_16x16x4_f32(false, a2, false, bq1, (short)0, acc1, false, false);
  }

  for (int r = 0; r < 8; ++r) {
    const int u = row0 + r + half * 8;
    for (int cc = 0; cc < 2; ++cc) {
      float t = alpha * (cc == 0 ? acc0[r] : acc1[r]);
      const long cg = cbase + c0 + cc * 16 + l15;
      const long o = obase + (long)u * ost + cg * osc;
      float res;
      if (mode == 0) {
        res = t;
      } else if (mode == 1) {
        // Zsum = 0.5*(X2*Ypl[v,c] + xh[..,fliproll(k,j)]*Ymi[v,c])
        const int k = (int)(cg >> 6), j = (int)(cg & 63);
        const long cp = (long)((64 - k) & 63) * 64 + (long)((64 - j) & 63);
        const float xf = xin[obase + (long)u * ost + cp * osc];
        const long yi = (long)v * 4096 + cg;
        res = 0.5f * (t * ypl[yi] + xf * ymi[yi]);
      } else if (mode == 2) {
        float tt = t + bias[cg];
        res = tt > 0.0f ? tt : 0.0f;
      } else if (mode == 3) {
        float tt = t + bias[cg];
        res = tt > lam ? tt - lam : (tt < -lam ? tt + lam : 0.0f);
      } else {
        res = t + xin[o];
      }
      out[o] = res;
    }
  }
}

// ---------------- weight combine: Ypl = Yt0+Yt1 + flip(w0)+flip(w1), Ymi = Yt - flips ----
__global__ void afno_wcomb(const float* __restrict__ yt0, const float* __restrict__ yt1,
                           const float* __restrict__ w0,  const float* __restrict__ w1r,
                           float* __restrict__ ypl, float* __restrict__ ymi)
{
  int i = blockIdx.x * blockDim.x + threadIdx.x;
  if (i >= 64 * 64 * 64) return;
  int m = i >> 12, a = (i >> 6) & 63, b = i & 63;
  int af = (64 - a) & 63, bf = (64 - b) & 63;
  int fi = (m << 12) + (af << 6) + bf;
  float yt = yt0[i] + yt1[i];
  float wf = w0[fi] + w1r[fi];
  ypl[i] = yt + wf;
  ymi[i] = yt - wf;
}

extern "C" void kernel_launch(void* const* d_in, const int* in_sizes, int n_in,
                              void* d_out, int out_size, void* d_ws, size_t ws_size,
                              hipStream_t stream) {
  (void)in_sizes; (void)n_in; (void)out_size; (void)ws_size;
  const float* x   = (const float*)d_in[0];   // (2, 4096, 4096)
  const float* w1p = (const float*)d_in[1];   // (2, 64, 64, 64)
  const float* b1p = (const float*)d_in[2];   // (2, 64, 64)
  const float* w2p = (const float*)d_in[3];   // (2, 64, 64, 64)
  const float* b2p = (const float*)d_in[4];   // (2, 64, 64)
  float* out = (float*)d_out;
  float* ws  = (float*)d_ws;

  // workspace layout (floats)
  const long O_CHAT = 0, O_SHAT = 4096, O_K1 = 8192, O_K2 = 12288;
  const long O_PW = 16384, O_QW = 278528;
  const long O_YT0 = 540672, O_YT1 = 802816;
  const long O_Y1PL = 1064960, O_Y1MI = 1327104, O_Y2PL = 1589248, O_Y2MI = 1851392;
  const long BIGN = 33554432L;            // 2*64*64*4096
  const long O_A = 2113536, O_B = O_A + BIGN, O_C = O_B + BIGN, O_D = O_C + BIGN;
  const long I30 = 1L << 30;

  afno_basis<<<16, 256, 0, stream>>>(ws + O_CHAT, ws + O_SHAT, ws + O_K1, ws + O_K2);

  // ---- weight spectra: Yt = dht2d(w[s]) per layer, then combine with flip-rolls ----
  for (int L = 0; L < 2; ++L) {
    const float* wb = (L == 0) ? w1p : w2p;
    for (int s = 0; s < 2; ++s) {
      const float* wsrc = wb + (long)s * 262144;
      // stage1 over last axis (stride 1), "c" = middle axis (stride 64)
      afno_g1<<<dim3(64, 1), 256, 0, stream>>>(wsrc, ws + O_PW, ws + O_QW,
          ws + O_CHAT, ws + O_SHAT,
          I30, 0L, 4096L, 1L, 64L,   I30, 0L, 4096L, 1L, 64L);
      // stage2 over middle axis (stride 64), "c" = freq axis (stride 1)
      afno_g2<<<dim3(64, 1), 256, 0, stream>>>(ws + O_PW, ws + O_QW,
          ws + (s == 0 ? O_YT0 : O_YT1), ws + O_K1, ws + O_K2,
          I30, 0L, 4096L, 64L, 1L,   I30, 0L, 4096L, 64L, 1L,
          1.0f / 64.0f, 0, nullptr, nullptr, nullptr, nullptr, 0.0f);
    }
    afno_wcomb<<<1024, 256, 0, stream>>>(ws + O_YT0, ws + O_YT1, wb, wb + 262144,
        ws + (L == 0 ? O_Y1PL : O_Y2PL), ws + (L == 0 ? O_Y1MI : O_Y2MI));
  }

  // ---- big DHT2 helpers: tensor layout (b,h|u, w|v, c) contiguous in c ----
  auto S1 = [&](const float* in) {
    afno_g1<<<dim3(128, 64), 256, 0, stream>>>(in, ws + O_A, ws + O_B,
        ws + O_CHAT, ws + O_SHAT,
        I30, 0L, 262144L, 4096L, 1L,   I30, 0L, 262144L, 4096L, 1L);
  };
  auto S2 = [&](float* dst, float alpha, int mode, const float* bias,
                const float* xin, const float* ypl, const float* ymi) {
    afno_g2<<<dim3(128, 64), 256, 0, stream>>>(ws + O_A, ws + O_B, dst,
        ws + O_K1, ws + O_K2,
        64L, 16777216L, 4096L, 262144L, 1L,   64L, 16777216L, 4096L, 262144L, 1L,
        alpha, mode, bias, xin, ypl, ymi, 0.01f);
  };

  const float INV64 = 1.0f / 64.0f;
  const float INV_IDHT = 4.6566128730773926e-10f;  // 1/(64 * 2*64^4) = 2^-31

  // 1) xh = dht2d(x)                                   -> slot C
  S1(x);                 S2(ws + O_C, INV64, 0, nullptr, nullptr, nullptr, nullptr);
  // 2) Zsum1 = 0.5*(dht2d(xh)*Y1pl + flip(xh)*Y1mi)    -> slot D
  S1(ws + O_C);          S2(ws + O_D, INV64, 1, nullptr, ws + O_C, ws + O_Y1PL, ws + O_Y1MI);
  // 3) o1 = relu(idht2d(Zsum1) + b1[0])                -> slot C
  S1(ws + O_D);          S2(ws + O_C, INV_IDHT, 2, b1p, nullptr, nullptr, nullptr);
  // 4) Zsum2 = 0.5*(dht2d(o1)*Y2pl + flip(o1)*Y2mi)    -> slot D
  S1(ws + O_C);          S2(ws + O_D, INV64, 1, nullptr, ws + O_C, ws + O_Y2PL, ws + O_Y2MI);
  // 5) z = softshrink(idht2d(Zsum2) + b2[0], 0.01)     -> slot C
  S1(ws + O_D);          S2(ws + O_C, INV_IDHT, 3, b2p, nullptr, nullptr, nullptr);
  // 6) out = idht2d(z) + x                             -> d_out
  S1(ws + O_C);          S2(out, INV_IDHT, 4, nullptr, x, nullptr, nullptr);
}